// SIBlocks_12232066859666
// MI455X (gfx1250) — compile-verified
//
#include <hip/hip_runtime.h>
#include <cstddef>

#define Bx     4
#define Nn     4096
#define Cc     64
#define KNEI   32
#define NKNOT  32
#define HG     64
#define EDGES  (Nn * KNEI)   /* 131072 */
#define RADIUS 0.2f
#define WIN    8

typedef float v2f __attribute__((ext_vector_type(2)));
typedef float v8f __attribute__((ext_vector_type(8)));

__device__ __forceinline__ float gcoord(int k) {
  // jnp.linspace(0,1,64): arange*step with endpoint forced to stop
  return (k == HG - 1) ? 1.0f : (float)k * (1.0f / 63.0f);
}
__device__ __forceinline__ float knot(int k) {
  // jnp.linspace(0,1,32)
  return (k == NKNOT - 1) ? 1.0f : (float)k * (1.0f / 31.0f);
}

// ---------------------------------------------------------------------------
// K0: zero the workspace accumulators (psi_sum @ ws[0], phi_colsum @ ws[1..64])
// ---------------------------------------------------------------------------
__global__ void k_zero(float* __restrict__ ws) {
  if (threadIdx.x < 80) ws[threadIdx.x] = 0.0f;
}

// ---------------------------------------------------------------------------
// K1: per-node top-32 neighbor selection (stable, matching jax.lax.top_k),
//     h = softplus(mlp2(ci, H1, hb1, H2, hb2)), psi = spline(r/(h+1e-6)),
//     and global sum of |psi|.
// ---------------------------------------------------------------------------
__global__ void k_nodes(const float* __restrict__ H1, const float* __restrict__ hb1,
                        const float* __restrict__ H2, const float* __restrict__ hb2,
                        const float* __restrict__ Sm, int* __restrict__ nbr,
                        float* __restrict__ psi_raw, float* __restrict__ psi_sum) {
  __shared__ float sd[64][KNEI];
  __shared__ int   sj[64][KNEI];
  __shared__ float red[64];
  const int t  = threadIdx.x;          // one node per thread
  const int i  = blockIdx.x * 64 + t;
  const int iy = i >> 6, ix = i & 63;
  const float cy = gcoord(iy), cx = gcoord(ix);

  #pragma unroll
  for (int k = 0; k < KNEI; ++k) { sd[t][k] = 3.0e38f; sj[t][k] = 0; }

  // Candidates scanned in ascending j (dy outer, dx inner). Stable insertion:
  // on ties the earlier-scanned (lower j) entry stays first -> matches top_k.
  for (int dy = -WIN; dy <= WIN; ++dy) {
    const int jy = iy + dy; if (jy < 0 || jy >= HG) continue;
    const float ddy = cy - gcoord(jy);
    for (int dx = -WIN; dx <= WIN; ++dx) {
      const int jx = ix + dx; if (jx < 0 || jx >= HG) continue;
      const float ddx = cx - gcoord(jx);
      const float d = sqrtf(ddy * ddy + ddx * ddx);
      if (d > RADIUS) continue;           // masked -> +inf, never selected
      if (d >= sd[t][KNEI - 1]) continue; // ties keep existing (lower j)
      int p = KNEI - 1;
      while (p > 0 && sd[t][p - 1] > d) {
        sd[t][p] = sd[t][p - 1]; sj[t][p] = sj[t][p - 1]; --p;
      }
      sd[t][p] = d; sj[t][p] = jy * HG + jx;
    }
  }

  // h-MLP: 2 -> 64 -> 1, relu, softplus (stable logaddexp form)
  float s = hb2[0];
  for (int ch = 0; ch < Cc; ++ch) {
    float tt = fmaf(cy, H1[ch], fmaf(cx, H1[Cc + ch], hb1[ch]));
    tt = fmaxf(tt, 0.0f);
    s  = fmaf(tt, H2[ch], s);
  }
  const float hval   = fmaxf(s, 0.0f) + log1pf(expf(-fabsf(s)));
  const float hdenom = hval + 1e-6f;

  float asum = 0.0f;
  for (int k = 0; k < KNEI; ++k) {
    float r = sd[t][k] / hdenom;
    r = fminf(fmaxf(r, 0.0f), 1.0f);
    int idx = (int)floorf(r * (float)(NKNOT - 1));
    idx = idx < 0 ? 0 : (idx > NKNOT - 2 ? NKNOT - 2 : idx);
    const float tk  = knot(idx);
    const float tk1 = knot(idx + 1);
    const float wr  = (r - tk) / (tk1 - tk + 1e-8f);
    const float ps  = (1.0f - wr) * Sm[idx] + wr * Sm[idx + 1];
    const int e = i * KNEI + k;
    nbr[e]     = sj[t][k];
    psi_raw[e] = ps;
    asum += fabsf(ps);
  }

  red[t] = asum;
  __syncthreads();
  for (int o = 32; o > 0; o >>= 1) { if (t < o) red[t] += red[t + o]; __syncthreads(); }
  if (t == 0) atomicAdd(psi_sum, red[0]);
}

// ---------------------------------------------------------------------------
// K2: phi = mlp2([ci,cj], P1, pb1, P2, pb2) on WMMA f32 16x16x4.
//     Layer 1 has K=4 exactly -> one WMMA per 16x16 tile.
//     Also accumulates per-channel sum of |phi|.
// ---------------------------------------------------------------------------
__global__ void k_phi(const float* __restrict__ P1, const float* __restrict__ pb1,
                      const float* __restrict__ P2, const float* __restrict__ pb2,
                      const int* __restrict__ nbr, float* __restrict__ phi_raw,
                      float* __restrict__ phi_colsum) {
  __shared__ float hid[8][16][68];   // 16x64 tile per wave, padded stride
  __shared__ float cs[64];
  const int lane = threadIdx.x & 31;
  const int wv   = threadIdx.x >> 5;
  const int half = lane >> 4;        // A/B K-pair select per ISA layout
  const int l16  = lane & 15;
  const int gw   = blockIdx.x * 8 + wv;
  const int nW   = gridDim.x * 8;
  if (threadIdx.x < 64) cs[threadIdx.x] = 0.0f;
  __syncthreads();

  float csacc[4] = {0.f, 0.f, 0.f, 0.f};
  for (int tile = gw; tile < EDGES / 16; tile += nW) {
    const int row = tile * 16 + l16;     // edge index of this lane's A row
    const int i   = row >> 5;            // row / KNEI
    const int j   = nbr[row];
    const float f0 = gcoord(i >> 6), f1 = gcoord(i & 63);
    const float f2 = gcoord(j >> 6), f3 = gcoord(j & 63);
    v2f a1; a1.x = half ? f2 : f0; a1.y = half ? f3 : f1;

    // layer 1: (16x4) @ (4x64), single WMMA per column tile
    for (int nt = 0; nt < 4; ++nt) {
      const int col = nt * 16 + l16;
      v2f bb;
      bb.x = P1[(half * 2 + 0) * Cc + col];
      bb.y = P1[(half * 2 + 1) * Cc + col];
      v8f acc = {};
      acc = __builtin_amdgcn_wmma_f32_16x16x4_f32(false, a1, false, bb,
                                                  (short)0, acc, false, false);
      const float bias = pb1[col];
      #pragma unroll
      for (int v = 0; v < 8; ++v)
        hid[wv][half * 8 + v][col] = fmaxf(acc[v] + bias, 0.0f);
    }
    __syncthreads();

    // layer 2: (16x64) @ (64x64), K in 16 steps of 4
    for (int nt = 0; nt < 4; ++nt) {
      const int col = nt * 16 + l16;
      v8f acc = {};
      #pragma unroll
      for (int kk = 0; kk < 16; ++kk) {
        const int K0 = kk * 4 + half * 2;
        v2f aa, bb;
        aa.x = hid[wv][l16][K0];
        aa.y = hid[wv][l16][K0 + 1];
        bb.x = P2[(K0 + 0) * Cc + col];
        bb.y = P2[(K0 + 1) * Cc + col];
        acc = __builtin_amdgcn_wmma_f32_16x16x4_f32(false, aa, false, bb,
                                                    (short)0, acc, false, false);
      }
      const float bias = pb2[col];
      float lsum = 0.0f;
      #pragma unroll
      for (int v = 0; v < 8; ++v) {
        const float val = acc[v] + bias;
        const int e = tile * 16 + half * 8 + v;
        phi_raw[(size_t)e * Cc + col] = val;
        lsum += fabsf(val);
      }
      csacc[nt] += lsum;
    }
    __syncthreads();
  }

  #pragma unroll
  for (int nt = 0; nt < 4; ++nt) atomicAdd(&cs[nt * 16 + l16], csacc[nt]);
  __syncthreads();
  if (threadIdx.x < 64) atomicAdd(&phi_colsum[threadIdx.x], cs[threadIdx.x]);
}

// ---------------------------------------------------------------------------
// K3: residual MLP  out = relu(x@W1 + b1)@W2 + b2  on WMMA f32 16x16x4.
//     16384 rows, 64 -> 128 -> 64.
// ---------------------------------------------------------------------------
__global__ void k_mlp(const float* __restrict__ x,  const float* __restrict__ W1,
                      const float* __restrict__ b1, const float* __restrict__ W2,
                      const float* __restrict__ b2, float* __restrict__ out) {
  __shared__ float hid[4][16][132];  // 16x128 tile per wave, padded stride
  const int lane = threadIdx.x & 31;
  const int wv   = threadIdx.x >> 5;
  const int half = lane >> 4;
  const int l16  = lane & 15;
  const int gw   = blockIdx.x * 4 + wv;
  const int nW   = gridDim.x * 4;
  const int ROWS = Bx * Nn;

  for (int tile = gw; tile < ROWS / 16; tile += nW) {
    const int row = tile * 16 + l16;
    v2f a1[16];
    #pragma unroll
    for (int kk = 0; kk < 16; ++kk)
      a1[kk] = *(const v2f*)(x + (size_t)row * Cc + kk * 4 + half * 2);

    // layer 1: K=64 (16 WMMA steps) per 16-col tile, 8 tiles of 128 cols
    for (int nt = 0; nt < 8; ++nt) {
      const int col = nt * 16 + l16;
      v8f acc = {};
      #pragma unroll
      for (int kk = 0; kk < 16; ++kk) {
        v2f bb;
        bb.x = W1[(size_t)(kk * 4 + half * 2 + 0) * 128 + col];
        bb.y = W1[(size_t)(kk * 4 + half * 2 + 1) * 128 + col];
        acc = __builtin_amdgcn_wmma_f32_16x16x4_f32(false, a1[kk], false, bb,
                                                    (short)0, acc, false, false);
      }
      const float bias = b1[col];
      #pragma unroll
      for (int v = 0; v < 8; ++v)
        hid[wv][half * 8 + v][col] = fmaxf(acc[v] + bias, 0.0f);
    }
    __syncthreads();

    // layer 2: K=128 (32 WMMA steps) per 16-col tile, 4 tiles of 64 cols
    for (int nt = 0; nt < 4; ++nt) {
      const int col = nt * 16 + l16;
      v8f acc = {};
      #pragma unroll
      for (int kk = 0; kk < 32; ++kk) {
        const int K0 = kk * 4 + half * 2;
        v2f aa, bb;
        aa.x = hid[wv][l16][K0];
        aa.y = hid[wv][l16][K0 + 1];
        bb.x = W2[(size_t)(K0 + 0) * Cc + col];
        bb.y = W2[(size_t)(K0 + 1) * Cc + col];
        acc = __builtin_amdgcn_wmma_f32_16x16x4_f32(false, aa, false, bb,
                                                    (short)0, acc, false, false);
      }
      const float bias = b2[col];
      #pragma unroll
      for (int v = 0; v < 8; ++v)
        out[(size_t)(tile * 16 + half * 8 + v) * Cc + col] = acc[v] + bias;
    }
    __syncthreads();
  }
}

// ---------------------------------------------------------------------------
// K4: out[b,i,c] += (1/32) * sum_k psi_n[e] * phi_n[e,c] * x[b, j(e), c]
// ---------------------------------------------------------------------------
__global__ void k_agg(const float* __restrict__ x, const int* __restrict__ nbr,
                      const float* __restrict__ psi_raw, const float* __restrict__ phi_raw,
                      const float* __restrict__ psi_sum, const float* __restrict__ phi_colsum,
                      float* __restrict__ out) {
  const int c = threadIdx.x;   // channel 0..63
  const int i = blockIdx.x;    // node
  const float pden = psi_sum[0]   / (float)EDGES + 1e-6f;
  const float fden = phi_colsum[c] / (float)EDGES + 1e-6f;

  float a0 = 0.f, a1 = 0.f, a2 = 0.f, a3 = 0.f;
  for (int k = 0; k < KNEI; ++k) {
    const int e = i * KNEI + k;
    const int j = nbr[e];
    const float w = (psi_raw[e] / pden) * (phi_raw[(size_t)e * Cc + c] / fden);
    const float* xp = x + (size_t)j * Cc + c;
    if (k + 1 < KNEI) __builtin_prefetch(x + (size_t)nbr[e + 1] * Cc + c, 0, 3);
    a0 = fmaf(w, xp[0],                      a0);
    a1 = fmaf(w, xp[(size_t)Nn * Cc],        a1);
    a2 = fmaf(w, xp[2 * (size_t)Nn * Cc],    a2);
    a3 = fmaf(w, xp[3 * (size_t)Nn * Cc],    a3);
  }
  const float s = 1.0f / 32.0f;
  const size_t o = (size_t)i * Cc + c;
  out[o]                        += a0 * s;
  out[(size_t)Nn * Cc + o]      += a1 * s;
  out[2 * (size_t)Nn * Cc + o]  += a2 * s;
  out[3 * (size_t)Nn * Cc + o]  += a3 * s;
}

// ---------------------------------------------------------------------------
extern "C" void kernel_launch(void* const* d_in, const int* in_sizes, int n_in,
                              void* d_out, int out_size, void* d_ws, size_t ws_size,
                              hipStream_t stream) {
  const float* x   = (const float*)d_in[0];
  const float* W1  = (const float*)d_in[1];
  const float* b1  = (const float*)d_in[2];
  const float* W2  = (const float*)d_in[3];
  const float* b2  = (const float*)d_in[4];
  const float* P1  = (const float*)d_in[5];
  const float* pb1 = (const float*)d_in[6];
  const float* P2  = (const float*)d_in[7];
  const float* pb2 = (const float*)d_in[8];
  const float* H1  = (const float*)d_in[9];
  const float* hb1 = (const float*)d_in[10];
  const float* H2  = (const float*)d_in[11];
  const float* hb2 = (const float*)d_in[12];
  const float* Sm  = (const float*)d_in[13];
  float* out = (float*)d_out;

  float* ws         = (float*)d_ws;
  float* psi_sum    = ws;                      // 1 float
  float* phi_colsum = ws + 1;                  // 64 floats
  int*   nbr        = (int*)(ws + 80);         // E ints
  float* psi_raw    = ws + 80 + EDGES;         // E floats
  float* phi_raw    = ws + 80 + 2 * EDGES;     // E*64 floats (~32 MB)

  hipLaunchKernelGGL(k_zero,  dim3(1),       dim3(128), 0, stream, ws);
  hipLaunchKernelGGL(k_nodes, dim3(Nn / 64), dim3(64),  0, stream,
                     H1, hb1, H2, hb2, Sm, nbr, psi_raw, psi_sum);
  hipLaunchKernelGGL(k_phi,   dim3(64),      dim3(256), 0, stream,
                     P1, pb1, P2, pb2, nbr, phi_raw, phi_colsum);
  hipLaunchKernelGGL(k_mlp,   dim3(64),      dim3(128), 0, stream,
                     x, W1, b1, W2, b2, out);
  hipLaunchKernelGGL(k_agg,   dim3(Nn),      dim3(64),  0, stream,
                     x, nbr, psi_raw, phi_raw, psi_sum, phi_colsum, out);
}